// AdaptiveSpectralFeatureRefinementCombined_72739566125924
// MI455X (gfx1250) — compile-verified
//
#include <hip/hip_runtime.h>
#include <math.h>

// Problem constants (fixed by the reference)
#define B_ 4
#define C_ 32
#define H_ 256
#define W_ 256
#define TILE 16
#define HALO (TILE + 2)      // 18: 16 + 1-halo each side
#define K_ 9                 // 3x3 taps
#define CSTRIDE 36           // per-pixel channel stride in LDS (padded, 144B, 16B-aligned)
#define NTHREADS 256
#define LDS_ELEMS (HALO * HALO * CSTRIDE)
#define EPS_ 1e-12f

// ---- CDNA5 async global->LDS path ------------------------------------------
#if defined(__has_builtin)
#if __has_builtin(__builtin_amdgcn_global_load_async_to_lds_b32)
#define HAVE_ASYNC_B32 1
#endif
#if __has_builtin(__builtin_amdgcn_s_wait_asynccnt)
#define HAVE_WAIT_ASYNC 1
#endif
#endif

#if !defined(HAVE_ASYNC_B32)
#warning "gfx1250 async-to-LDS builtin NOT available; falling back to ld/st staging"
#endif

typedef __attribute__((address_space(1))) int* gptr_i32;
typedef __attribute__((address_space(3))) int* lptr_i32;

__device__ __forceinline__ void stage_b32(const float* g, float* l) {
#if defined(HAVE_ASYNC_B32)
  // global_load_async_to_lds_b32: per-lane global addr -> per-lane LDS addr,
  // tracked by ASYNCcnt (no VGPR data staging).
  __builtin_amdgcn_global_load_async_to_lds_b32(
      (gptr_i32)g, (lptr_i32)l, /*offset=*/0, /*cpol=*/0);
#else
  *l = *g;
#endif
}

__device__ __forceinline__ void drain_async() {
#if defined(HAVE_ASYNC_B32)
#if defined(HAVE_WAIT_ASYNC)
  __builtin_amdgcn_s_wait_asynccnt(0);
#else
  asm volatile("s_wait_asynccnt 0" ::: "memory");
#endif
#endif
}
// ---------------------------------------------------------------------------

__global__ __launch_bounds__(NTHREADS) void asfr_kernel(
    const float* __restrict__ fe, const float* __restrict__ fused,
    float* __restrict__ out) {
  __shared__ __align__(16) float lds[LDS_ELEMS];  // [y][x][c] with padded c-stride

  const int tid = threadIdx.x;
  const int b = blockIdx.z;
  const int baseX = blockIdx.x * TILE;
  const int baseY = blockIdx.y * TILE;
  const size_t plane = (size_t)H_ * W_;

  const float* __restrict__ fusedB = fused + (size_t)b * C_ * plane;
  const float* __restrict__ feB = fe + (size_t)b * C_ * plane;
  float* __restrict__ outB = out + (size_t)b * C_ * plane;

  // Zero-padding only needed for tiles touching the image border (uniform branch).
  const bool border = (baseX == 0) || (baseY == 0) ||
                      (baseX + TILE == W_) || (baseY + TILE == H_);
  if (border) {
    for (int e = tid; e < LDS_ELEMS; e += NTHREADS) lds[e] = 0.0f;
    __syncthreads();
  }

  // Stage fused halo tile into LDS, transposing [c][y][x] -> [y][x][c].
  // e is ordered x-fastest within a (c,y) row so lanes read contiguous global
  // addresses; the LDS side is a per-lane scatter handled by the async unit.
  for (int e = tid; e < C_ * HALO * HALO; e += NTHREADS) {
    const int x = e % HALO;
    const int t = e / HALO;
    const int y = t % HALO;
    const int c = t / HALO;
    const int gx = baseX + x - 1;
    const int gy = baseY + y - 1;
    if (((unsigned)gx < (unsigned)W_) && ((unsigned)gy < (unsigned)H_)) {
      stage_b32(fusedB + (size_t)c * plane + (size_t)gy * W_ + gx,
                &lds[(y * HALO + x) * CSTRIDE + c]);
    }
  }
  drain_async();
  __syncthreads();

  // ---- per-pixel compute ----
  const int px = tid & (TILE - 1);
  const int py = tid >> 4;
  const size_t pixOff = (size_t)(baseY + py) * W_ + (size_t)(baseX + px);

  // fe channel vector (coalesced per channel plane) + its norm
  float fev[C_];
  float fn2 = 0.0f;
#pragma unroll
  for (int c = 0; c < C_; ++c) {
    const float v = feB[(size_t)c * plane + pixOff];
    fev[c] = v;
    fn2 = fmaf(v, v, fn2);
  }
  const float fnorm = fmaxf(sqrtf(fn2), EPS_);

  // Pass 1 over the 9 taps: cosine sim + (negative) euclidean distance
  float cosv[K_], negd[K_];
#pragma unroll
  for (int k = 0; k < K_; ++k) {
    const float* P = &lds[((py + k / 3) * HALO + (px + k % 3)) * CSTRIDE];
    float dot = 0.0f, pn2 = 0.0f, d2 = 0.0f;
#pragma unroll
    for (int c = 0; c < C_; c += 4) {
      const float4 p = *(const float4*)(P + c);
      dot = fmaf(p.x, fev[c + 0], dot);
      dot = fmaf(p.y, fev[c + 1], dot);
      dot = fmaf(p.z, fev[c + 2], dot);
      dot = fmaf(p.w, fev[c + 3], dot);
      pn2 = fmaf(p.x, p.x, pn2);
      pn2 = fmaf(p.y, p.y, pn2);
      pn2 = fmaf(p.z, p.z, pn2);
      pn2 = fmaf(p.w, p.w, pn2);
      const float dx0 = p.x - fev[c + 0];
      const float dx1 = p.y - fev[c + 1];
      const float dx2 = p.z - fev[c + 2];
      const float dx3 = p.w - fev[c + 3];
      d2 = fmaf(dx0, dx0, d2);
      d2 = fmaf(dx1, dx1, d2);
      d2 = fmaf(dx2, dx2, d2);
      d2 = fmaf(dx3, dx3, d2);
    }
    cosv[k] = dot / (fmaxf(sqrtf(pn2), EPS_) * fnorm);
    negd[k] = -sqrtf(d2);
  }

  // Two 9-way softmaxes, blended 0.5/0.5
  float mc = cosv[0], md = negd[0];
#pragma unroll
  for (int k = 1; k < K_; ++k) {
    mc = fmaxf(mc, cosv[k]);
    md = fmaxf(md, negd[k]);
  }
  float ec[K_], ed[K_];
  float sc = 0.0f, sd = 0.0f;
#pragma unroll
  for (int k = 0; k < K_; ++k) {
    ec[k] = expf(cosv[k] - mc);
    sc += ec[k];
    ed[k] = expf(negd[k] - md);
    sd += ed[k];
  }
  const float isc = 0.5f / sc;
  const float isd = 0.5f / sd;
  float w[K_];
#pragma unroll
  for (int k = 0; k < K_; ++k) w[k] = fmaf(ec[k], isc, ed[k] * isd);

  // Pass 2: weighted accumulation; start from fe (final "+ fe_lv" is free)
  float acc[C_];
#pragma unroll
  for (int c = 0; c < C_; ++c) acc[c] = fev[c];
#pragma unroll
  for (int k = 0; k < K_; ++k) {
    const float* P = &lds[((py + k / 3) * HALO + (px + k % 3)) * CSTRIDE];
    const float wk = w[k];
#pragma unroll
    for (int c = 0; c < C_; c += 4) {
      const float4 p = *(const float4*)(P + c);
      acc[c + 0] = fmaf(wk, p.x, acc[c + 0]);
      acc[c + 1] = fmaf(wk, p.y, acc[c + 1]);
      acc[c + 2] = fmaf(wk, p.z, acc[c + 2]);
      acc[c + 3] = fmaf(wk, p.w, acc[c + 3]);
    }
  }

  // Coalesced stores (consecutive px across lanes within each channel plane)
#pragma unroll
  for (int c = 0; c < C_; ++c) outB[(size_t)c * plane + pixOff] = acc[c];
}

extern "C" void kernel_launch(void* const* d_in, const int* in_sizes, int n_in,
                              void* d_out, int out_size, void* d_ws,
                              size_t ws_size, hipStream_t stream) {
  (void)in_sizes; (void)n_in; (void)out_size; (void)d_ws; (void)ws_size;
  const float* fe = (const float*)d_in[0];     // fe_lv
  const float* fused = (const float*)d_in[1];  // fused_features
  float* out = (float*)d_out;

  dim3 grid(W_ / TILE, H_ / TILE, B_);  // (16,16,4) = 1024 workgroups
  asfr_kernel<<<grid, NTHREADS, 0, stream>>>(fe, fused, out);
}